// TensorDecomposition_42614665511551
// MI455X (gfx1250) — compile-verified
//
#include <hip/hip_runtime.h>
#include <hip/hip_bf16.h>

#define GRID_N   512
#define F_FEAT   32
#define N_PTS    2097152   // 2^21
#define PLANE_TEXELS (GRID_N * GRID_N)                 // 262144
#define PLANE_FLOATS ((size_t)PLANE_TEXELS * F_FEAT)   // 8388608 floats per plane
#define PAD_FLOATS   32768                             // 128 KiB zero guard after planes

typedef __attribute__((ext_vector_type(2))) float v2f;
typedef __attribute__((ext_vector_type(8))) float v8f;

// ---------------------------------------------------------------------------
// Kernel 1: transpose (F, H, W) channel-major planes into (H, W, F) point-major
// layout in workspace, so each bilinear corner is one contiguous 128B line.
// Source reads are NT (read-once; don't pollute L2 — the transposed copy is
// the one we want resident).
// ---------------------------------------------------------------------------
__global__ __launch_bounds__(256)
void tp_transpose_kernel(const float* __restrict__ pxy,
                         const float* __restrict__ pyz,
                         const float* __restrict__ pxz,
                         float* __restrict__ dst) {
    const float* src = (blockIdx.y == 0) ? pxy : ((blockIdx.y == 1) ? pyz : pxz);
    const int texel = blockIdx.x * blockDim.x + threadIdx.x;   // 0 .. 262143
    float v[F_FEAT];
#pragma unroll
    for (int f = 0; f < F_FEAT; ++f)
        v[f] = __builtin_nontemporal_load(&src[(size_t)f * PLANE_TEXELS + texel]);
    float4* o = (float4*)(dst + ((size_t)blockIdx.y * PLANE_TEXELS + texel) * F_FEAT);
#pragma unroll
    for (int j = 0; j < F_FEAT / 4; ++j)
        o[j] = make_float4(v[4*j], v[4*j+1], v[4*j+2], v[4*j+3]);
}

// Zero the guard region past the planes (edge corners with weight 0 read here;
// must be finite so fma(0, x, acc) == acc).
__global__ __launch_bounds__(256)
void tp_zero_pad(float* __restrict__ pad) {
    pad[blockIdx.x * blockDim.x + threadIdx.x] = 0.0f;
}

// ---------------------------------------------------------------------------
// Kernel 2: scale[f] = sum_k core[k, f]  ==  ones(1x16) @ core(16x32)
// via V_WMMA_F32_16X16X4_F32 with accumulator chaining (4 K-chunks x 2 N-halves).
// A = all-ones is invariant under the K-row striping, so the sum is exact fp32.
// ---------------------------------------------------------------------------
__global__ __launch_bounds__(32)
void tp_core_scale_wmma(const float* __restrict__ core, float* __restrict__ scale) {
    const int lane = threadIdx.x;           // one wave32
    v2f a; a.x = 1.0f; a.y = 1.0f;          // A(16x4) = all ones

    for (int half = 0; half < 2; ++half) {  // N columns 0..15 then 16..31
        const int n = (lane & 15) + half * 16;
        const int kb = (lane < 16) ? 0 : 2; // B 4x16: lanes 16-31 hold K=2,3
        v8f c = {};
#pragma unroll
        for (int k0 = 0; k0 < 16; k0 += 4) {
            v2f b;
            b.x = core[(k0 + kb + 0) * F_FEAT + n];
            b.y = core[(k0 + kb + 1) * F_FEAT + n];
            c = __builtin_amdgcn_wmma_f32_16x16x4_f32(
                    false, a, false, b, (short)0, c, false, false);
        }
        if (lane < 16) scale[n] = c[0];     // M=0 row, VGPR0, lanes 0..15
    }
}

// ---------------------------------------------------------------------------
// Kernel 3: main sampler.  lane == feature channel; each wave owns 32 points.
// Per lane: precompute (off00, wx, wy) for its own point per plane (coalesced
// NT coord loads), then iterate the 32 points broadcasting just 3 scalars per
// plane via v_readlane (uniform index -> SGPR).  The 4 corner loads per plane
// share one scalar base address with immediate offsets and a lane*4 voffset:
// fully coalesced 128B L2-resident transactions.  Output stores are NT so the
// 256MB stream never evicts the planes from L2.
// ---------------------------------------------------------------------------
__global__ __launch_bounds__(256)
void tp_sample_kernel(const float* __restrict__ coords,
                      const float* __restrict__ planes_t,   // (3, H, W, F) + pad
                      const float* __restrict__ scale_v,    // (32,)
                      float* __restrict__ out) {
    const int lane   = threadIdx.x & 31;
    const int wave   = blockIdx.x * (blockDim.x >> 5) + (threadIdx.x >> 5);
    const int ptBase = wave * 32;
    const int myPt   = ptBase + lane;

    const float fscale = scale_v[lane];

    int   off00[3];
    float wxv[3], wyv[3];
#pragma unroll
    for (int p = 0; p < 3; ++p) {
        const size_t cbase = (size_t)p * N_PTS * 2 + (size_t)myPt * 2;
        const float cx = __builtin_nontemporal_load(&coords[cbase + 0]);
        const float cy = __builtin_nontemporal_load(&coords[cbase + 1]);
        const float x = (cx + 1.0f) * 0.5f * (GRID_N - 1);
        const float y = (cy + 1.0f) * 0.5f * (GRID_N - 1);
        const float x0f = floorf(x), y0f = floorf(y);
        wxv[p] = x - x0f;                   // in [0,1); == 0 at the x==511 edge
        wyv[p] = y - y0f;
        const int x0 = (int)x0f;            // coords in [-1,1] => x in [0,511]
        const int y0 = (int)y0f;
        off00[p] = (p * PLANE_TEXELS + y0 * GRID_N + x0) * F_FEAT;
    }

#pragma unroll 2
    for (int i = 0; i < 32; ++i) {
        float prod = fscale;
#pragma unroll
        for (int p = 0; p < 3; ++p) {
            const int   o  = __builtin_amdgcn_readlane(off00[p], i);      // SGPR
            const float wx = __int_as_float(
                                 __builtin_amdgcn_readlane(__float_as_int(wxv[p]), i));
            const float wy = __int_as_float(
                                 __builtin_amdgcn_readlane(__float_as_int(wyv[p]), i));
            const float* bp = planes_t + o;            // uniform base (saddr)
            const float v00 = bp[lane];                                   // +0
            const float v01 = bp[lane + F_FEAT];                          // +128B
            const float v10 = bp[lane + GRID_N * F_FEAT];                 // +64KiB
            const float v11 = bp[lane + GRID_N * F_FEAT + F_FEAT];        // +64KiB+128B
            const float ix = 1.0f - wx, iy = 1.0f - wy;
            float acc;
            acc = ix * iy * v00;
            acc = fmaf(wx * iy, v01, acc);
            acc = fmaf(ix * wy, v10, acc);
            acc = fmaf(wx * wy, v11, acc);
            prod *= acc;
        }
        __builtin_nontemporal_store(prod, &out[(size_t)(ptBase + i) * F_FEAT + lane]);
    }
}

// ---------------------------------------------------------------------------
// Fallback (workspace too small): same wave mapping, gathers from the original
// channel-major layout (lane f reads stride-1MB apart).  Correct, slower.
// ---------------------------------------------------------------------------
__global__ __launch_bounds__(256)
void tp_sample_fallback(const float* __restrict__ coords,
                        const float* __restrict__ pxy,
                        const float* __restrict__ pyz,
                        const float* __restrict__ pxz,
                        const float* __restrict__ core,
                        float* __restrict__ out) {
    const int lane   = threadIdx.x & 31;
    const int wave   = blockIdx.x * (blockDim.x >> 5) + (threadIdx.x >> 5);
    const int ptBase = wave * 32;
    const int myPt   = ptBase + lane;

    float fscale = 0.0f;
#pragma unroll
    for (int k = 0; k < 16; ++k) fscale += core[k * F_FEAT + lane];

    const float* planes[3] = {pxy, pyz, pxz};
    int   off[3][4];
    float w[3][4];
#pragma unroll
    for (int p = 0; p < 3; ++p) {
        const size_t cbase = (size_t)p * N_PTS * 2 + (size_t)myPt * 2;
        const float cx = coords[cbase + 0];
        const float cy = coords[cbase + 1];
        const float x = (cx + 1.0f) * 0.5f * (GRID_N - 1);
        const float y = (cy + 1.0f) * 0.5f * (GRID_N - 1);
        const float x0f = floorf(x), y0f = floorf(y);
        const float wx = x - x0f,    wy = y - y0f;
        int x0 = min(max((int)x0f, 0), GRID_N - 1);
        int y0 = min(max((int)y0f, 0), GRID_N - 1);
        int x1 = min(x0 + 1, GRID_N - 1);
        int y1 = min(y0 + 1, GRID_N - 1);
        off[p][0] = y0 * GRID_N + x0;
        off[p][1] = y0 * GRID_N + x1;
        off[p][2] = y1 * GRID_N + x0;
        off[p][3] = y1 * GRID_N + x1;
        w[p][0] = (1.0f - wx) * (1.0f - wy);
        w[p][1] = wx * (1.0f - wy);
        w[p][2] = (1.0f - wx) * wy;
        w[p][3] = wx * wy;
    }

    const size_t lbase = (size_t)lane * PLANE_TEXELS;
    for (int i = 0; i < 32; ++i) {
        float prod = fscale;
#pragma unroll
        for (int p = 0; p < 3; ++p) {
            float acc = 0.0f;
#pragma unroll
            for (int c = 0; c < 4; ++c) {
                const int   o  = __builtin_amdgcn_readlane(off[p][c], i);
                const float ww = __int_as_float(
                                     __builtin_amdgcn_readlane(__float_as_int(w[p][c]), i));
                acc = fmaf(ww, planes[p][lbase + o], acc);
            }
            prod *= acc;
        }
        __builtin_nontemporal_store(prod, &out[(size_t)(ptBase + i) * F_FEAT + lane]);
    }
}

extern "C" void kernel_launch(void* const* d_in, const int* in_sizes, int n_in,
                              void* d_out, int out_size, void* d_ws, size_t ws_size,
                              hipStream_t stream) {
    const float* coords = (const float*)d_in[0];  // (3,1,N,1,2)
    const float* pxy    = (const float*)d_in[1];  // (1,32,512,512)
    const float* pyz    = (const float*)d_in[2];
    const float* pxz    = (const float*)d_in[3];
    const float* core   = (const float*)d_in[4];  // (16,32)
    float*       out    = (float*)d_out;          // (N,32)

    const size_t planesBytes = 3 * PLANE_FLOATS * sizeof(float);      // 96 MB
    const size_t padBytes    = (size_t)PAD_FLOATS * sizeof(float);    // 128 KiB
    const size_t needBytes   = planesBytes + padBytes + 256;

    const int totalWaves = N_PTS / 32;                 // 65536
    const int blocks     = totalWaves / (256 / 32);    // 8192

    if (ws_size >= needBytes) {
        float* planes_t = (float*)d_ws;
        float* pad      = (float*)((char*)d_ws + planesBytes);
        float* scale_v  = (float*)((char*)d_ws + planesBytes + padBytes);

        dim3 tgrid(PLANE_TEXELS / 256, 3);
        tp_transpose_kernel<<<tgrid, 256, 0, stream>>>(pxy, pyz, pxz, planes_t);
        tp_zero_pad<<<PAD_FLOATS / 256, 256, 0, stream>>>(pad);
        tp_core_scale_wmma<<<1, 32, 0, stream>>>(core, scale_v);
        tp_sample_kernel<<<blocks, 256, 0, stream>>>(coords, planes_t, scale_v, out);
    } else {
        tp_sample_fallback<<<blocks, 256, 0, stream>>>(coords, pxy, pyz, pxz, core, out);
    }
}